// Linear_88751204204632
// MI455X (gfx1250) — compile-verified
//
#include <hip/hip_runtime.h>
#include <hip/hip_bf16.h>

// ArcFace-style margin head:
//   out = S * cosine                         (everywhere)
//   out[r, label[r]] = S * (-A*acos(c) + B)  (iff label[r] != -1)
//
// Memory-bound: ~410 MB of traffic -> ~17.6 us floor at 23.3 TB/s HBM.
// Strategy: pure streaming kernel with b128 non-temporal load/store
// (input alone is 204.8 MB > 192 MB L2, each byte touched once), packed
// f32 math, wave32 blocks of 256 (8 waves). Sparse per-row margin fixup
// as a second tiny kernel on the same stream.

#define MARGIN_A 0.88f
#define MARGIN_B 0.88f
#define SCALE_S  64.0f

typedef float v4f __attribute__((ext_vector_type(4)));

// ---------------------------------------------------------------------------
// Kernel 1: streaming scale, branch-free hot path. One float4 per thread ->
// global_load_b128 th:TH_LOAD_NT, 2x v_pk_mul_f32, global_store_b128
// th:TH_STORE_NT.
// ---------------------------------------------------------------------------
__global__ __launch_bounds__(256) void scale_stream(const float* __restrict__ in,
                                                    float* __restrict__ out,
                                                    long long n4) {  // # of float4 vectors
    long long i = (long long)blockIdx.x * blockDim.x + threadIdx.x;
    if (i < n4) {
        const v4f* vin  = (const v4f*)in;
        v4f*       vout = (v4f*)out;
        v4f v = __builtin_nontemporal_load(vin + i);
        v *= SCALE_S;
        __builtin_nontemporal_store(v, vout + i);
    }
}

// ---------------------------------------------------------------------------
// Kernel 2: per-row target fixup. 512 rows -> 512 threads. Reads the pristine
// input (never mutated), so it is idempotent across graph replays. Ordered
// after scale_stream by the stream.
// ---------------------------------------------------------------------------
__global__ __launch_bounds__(256) void target_fixup(const float* __restrict__ cosine,
                                                    const int* __restrict__ label,
                                                    float* __restrict__ out,
                                                    int batch,
                                                    long long num_classes) {
    int r = blockIdx.x * blockDim.x + threadIdx.x;
    if (r >= batch) return;
    int lbl = label[r];
    if (lbl == -1) return;               // pass-through rows: already S*cosine
    long long idx = (long long)r * num_classes + (long long)lbl;
    float c = cosine[idx];
    out[idx] = SCALE_S * (fmaf(-MARGIN_A, acosf(c), MARGIN_B));
}

// ---------------------------------------------------------------------------
// Kernel 3: scalar tail (<= 3 elements). Only launched when n_total % 4 != 0,
// which never happens for this problem shape (512*100000), so the timed graph
// contains just the two kernels above.
// ---------------------------------------------------------------------------
__global__ void scale_tail(const float* __restrict__ in,
                           float* __restrict__ out,
                           long long start, long long n_total) {
    long long t = start + threadIdx.x;
    if (t < n_total) out[t] = SCALE_S * in[t];
}

extern "C" void kernel_launch(void* const* d_in, const int* in_sizes, int n_in,
                              void* d_out, int out_size, void* d_ws, size_t ws_size,
                              hipStream_t stream) {
    (void)n_in; (void)out_size; (void)d_ws; (void)ws_size;

    const float* cosine = (const float*)d_in[0];
    const int*   label  = (const int*)d_in[1];
    float*       out    = (float*)d_out;

    const long long n_total     = (long long)in_sizes[0];      // 512 * 100000
    const int       batch       = in_sizes[1];                 // 512
    const long long num_classes = n_total / (long long)batch;  // 100000

    // Pass 1: full streaming scale (writes every vector-covered element).
    const long long n4 = n_total / 4;
    const int block = 256;
    const long long grid1 = (n4 + block - 1) / block;
    scale_stream<<<(unsigned)grid1, block, 0, stream>>>(cosine, out, n4);

    // Tail (shape-generality only; not launched for this problem size).
    if (n_total % 4 != 0) {
        scale_tail<<<1, 4, 0, stream>>>(cosine, out, n4 * 4, n_total);
    }

    // Pass 2: overwrite one element per valid row with the margin value.
    const int grid2 = (batch + block - 1) / block;
    target_fixup<<<grid2, block, 0, stream>>>(cosine, label, out, batch, num_classes);
}